// PeriodGuidedMultiScaleEmbeding_90675349553319
// MI455X (gfx1250) — compile-verified
//
#include <hip/hip_runtime.h>
#include <math.h>
#include <stdint.h>

// ---------------------------------------------------------------------------
// PeriodGuidedMultiScaleEmbeding for MI455X (gfx1250), fp32 WMMA path.
// Heavy GEMMs: V_WMMA_F32_16X16X4_F32, B panels staged into LDS with
// GLOBAL_LOAD_ASYNC_TO_LDS_B32 (ASYNCcnt) shared by all 8 waves of a block.
// ---------------------------------------------------------------------------

typedef float v2f __attribute__((ext_vector_type(2)));
typedef float v8f __attribute__((ext_vector_type(8)));

#define NEXP 17
#define SEQ  96
#define NB   64
#define DE   16
#define NF   48
#define OUT_STRIDE 26736            // D * sum(PAD_LENS) = 16*1671
#define OUT_MAIN   (NB * OUT_STRIDE)

__device__ __forceinline__ v8f wmma_f32(v2f a, v2f b, v8f c) {
    return __builtin_amdgcn_wmma_f32_16x16x4_f32(
        false, a, false, b, (short)0, c, false, false);
}

__device__ __forceinline__ void wait_asynccnt0() {
#if __has_builtin(__builtin_amdgcn_s_wait_asynccnt)
    __builtin_amdgcn_s_wait_asynccnt(0);
#else
    asm volatile("s_wait_asynccnt 0x0" ::: "memory");
#endif
}

// ---------------------------------------------------------------------------
// Sinusoid positional-encoding table: pe[t][d], t < 192, d < 16
// ---------------------------------------------------------------------------
__global__ __launch_bounds__(256) void pe_kernel(float* __restrict__ pe) {
    int idx = blockIdx.x * blockDim.x + threadIdx.x;
    if (idx >= 192 * DE) return;
    int t = idx >> 4, d = idx & 15;
    int i = d >> 1;
    float div = __expf((2.0f * (float)i) * (-logf(10000.0f) / (float)DE));
    float v = (float)t * div;
    pe[idx] = (d & 1) ? __cosf(v) : __sinf(v);
}

// ---------------------------------------------------------------------------
// Gating: per-batch AFNO1D gate.  One block of 96 threads per batch.
// ---------------------------------------------------------------------------
__global__ __launch_bounds__(96) void gating_kernel(
    const float* __restrict__ x, const float* __restrict__ start_w,
    const float* __restrict__ start_b,
    const float* __restrict__ w1, const float* __restrict__ b1,
    const float* __restrict__ w2, const float* __restrict__ b2,
    const float* __restrict__ w_gate,
    float* __restrict__ gates, float* __restrict__ out_tail)
{
    __shared__ float s[SEQ];
    __shared__ float xr[NF], xi[NF];
    __shared__ float o1r[NF * 4], o1i[NF * 4];
    __shared__ float amp[NF];
    __shared__ float logits[NEXP];

    const int b = blockIdx.x;
    const int t = threadIdx.x;

    {
        const float* xb = x + ((long)b * SEQ + t) * 32;
        float acc = 0.f;
        #pragma unroll
        for (int c = 0; c < 32; ++c) acc += xb[c] * start_w[c];
        s[t] = acc + start_b[0];
    }
    __syncthreads();

    if (t < NF) {                          // rfft (ortho), drop DC
        int k = t + 1;
        float w = -2.0f * 3.14159265358979323846f * (float)k / (float)SEQ;
        float sr = 0.f, si = 0.f;
        for (int u = 0; u < SEQ; ++u) {
            float ang = w * (float)u;
            sr += s[u] * __cosf(ang);
            si += s[u] * __sinf(ang);
        }
        const float sc = 0.102062072615966f;   // 1/sqrt(96)
        xr[t] = sr * sc;
        xi[t] = si * sc;
    }
    __syncthreads();

    for (int n = t; n < NF * 4; n += 96) {  // complex layer 1 + relu
        float ar = b1[n], ai = b1[NF * 4 + n];
        for (int k = 0; k < NF; ++k) {
            float w0 = w1[k * (NF * 4) + n];
            float wI = w1[NF * (NF * 4) + k * (NF * 4) + n];
            float r = xr[k], im = xi[k];
            ar += r * w0 - im * wI;
            ai += im * w0 + r * wI;
        }
        o1r[n] = fmaxf(ar, 0.f);
        o1i[n] = fmaxf(ai, 0.f);
    }
    __syncthreads();

    if (t < NF) {                           // complex layer 2 + amplitude
        float ar = b2[t], ai = b2[NF + t];
        for (int n = 0; n < NF * 4; ++n) {
            float w0 = w2[n * NF + t];
            float wI = w2[(NF * 4) * NF + n * NF + t];
            float r = o1r[n], im = o1i[n];
            ar += r * w0 - im * wI;
            ai += im * w0 + r * wI;
        }
        amp[t] = sqrtf(ar * ar + ai * ai);
    }
    __syncthreads();

    if (t < NEXP) {
        float l = 0.f;
        for (int k = 0; k < NF; ++k) l += amp[k] * w_gate[k * NEXP + t];
        logits[t] = l;
    }
    __syncthreads();

    if (t == 0) {                           // top-2 + softmax
        int i0 = 0; float l0 = logits[0];
        for (int e = 1; e < NEXP; ++e) if (logits[e] > l0) { l0 = logits[e]; i0 = e; }
        int i1 = (i0 == 0) ? 1 : 0; float l1 = logits[i1];
        for (int e = 0; e < NEXP; ++e)
            if (e != i0 && logits[e] > l1) { l1 = logits[e]; i1 = e; }
        float mx = fmaxf(l0, l1);
        float e0 = __expf(l0 - mx), e1 = __expf(l1 - mx);
        float inv = 1.f / (e0 + e1);
        for (int e = 0; e < NEXP; ++e) {
            float g = (e == i0) ? e0 * inv : ((e == i1) ? e1 * inv : 0.f);
            gates[b * NEXP + e] = g;
            out_tail[b * NEXP + e] = g;
        }
    }
}

// ---------------------------------------------------------------------------
// Simple fp32 WMMA GEMM (one 16x16 tile per wave) — used for the embedding
// (N = 16).  Row m -> (b = m/rowsPerB, f = m%rowsPerB).
// ---------------------------------------------------------------------------
__global__ __launch_bounds__(256) void gemm_wmma_nt1(
    int M, int N, int K,
    const float* __restrict__ A, long aStrideB, long aStrideF, int rowsPerB,
    const float* __restrict__ Bw,
    const float* __restrict__ bias,
    float* __restrict__ C, long cStrideB, long cStrideF,
    const float* __restrict__ gates, int expert, int relu)
{
    const int lane = threadIdx.x & 31;
    const int wid  = blockIdx.x * (blockDim.x >> 5) + (threadIdx.x >> 5);
    const int tilesN = N >> 4;
    const int tilesM = M >> 4;
    if (wid >= tilesM * tilesN) return;

    const int tm = wid / tilesN;
    const int tn = wid - tm * tilesN;
    const int lo = lane & 15;
    const int hi = lane >> 4;

    const int m = tm * 16 + lo;
    const float* Arow = A + (long)(m / rowsPerB) * aStrideB
                          + (long)(m % rowsPerB) * aStrideF;
    const float* Bcol = Bw + tn * 16 + lo;

    v8f c = {};
    for (int k0 = 0; k0 < K; k0 += 4) {
        const int ka = k0 + 2 * hi;          // reg j -> K = k0 + j + 2*hi
        v2f a, bv;
        a.x  = Arow[ka];
        a.y  = Arow[ka + 1];
        bv.x = Bcol[(long)ka * N];
        bv.y = Bcol[(long)(ka + 1) * N];
        c = wmma_f32(a, bv, c);
    }

    const int col = tn * 16 + lo;
    const float bval = bias ? bias[col] : 0.f;
    #pragma unroll
    for (int r = 0; r < 8; ++r) {
        const int row = tm * 16 + r + 8 * hi;   // C reg r -> row r + 8*hi
        const int bb = row / rowsPerB;
        const int ff = row - bb * rowsPerB;
        float v = c[r] + bval;
        if (relu) v = fmaxf(v, 0.f);
        if (gates) v *= (gates[bb * NEXP + expert] > 0.f) ? 1.f : 0.f;
        C[(long)bb * cStrideB + (long)ff * cStrideF + col] = v;
    }
}

// ---------------------------------------------------------------------------
// Main fp32 WMMA GEMM for the expert MLPs.
//   Block = 8 waves.  Block computes a 128-row M panel x 32-col N panel.
//   Wave w: M tile (pm*8+w), both 16-col N tiles of the panel (A reuse,
//   two independent accumulator chains).
//   B panel (8 K x 32 N) is staged into LDS once per block per K-chunk with
//   GLOBAL_LOAD_ASYNC_TO_LDS_B32 (ASYNCcnt), then read by all waves via DS.
// Requirements (guaranteed by problem dims): M%16==0, N%16==0, K%8==0.
// Ragged N (odd tilesN) handled by a uniform per-block second-tile guard.
// ---------------------------------------------------------------------------
__global__ __launch_bounds__(256) void gemm_wmma_nt2(
    int M, int N, int K,
    const float* __restrict__ A, long aStrideB, long aStrideF, int rowsPerB,
    const float* __restrict__ Bw,
    const float* __restrict__ bias,
    float* __restrict__ C, long cStrideB, long cStrideF,
    const float* __restrict__ gates, int expert, int relu, int panelsN)
{
    __shared__ float Bs[8][32];             // K-chunk x N-panel, LDS offset 0

    const int tid  = threadIdx.x;
    const int wave = tid >> 5;
    const int lane = tid & 31;
    const int lo   = lane & 15;
    const int hi   = lane >> 4;

    const int pn = blockIdx.x % panelsN;
    const int pm = blockIdx.x / panelsN;

    const int tilesM = M >> 4;
    const int tilesN = N >> 4;
    const int tm     = pm * 8 + wave;
    const bool active = (tm < tilesM);
    const bool nt1ok  = (pn * 2 + 1) < tilesN;
    const int colBase = pn * 32;

    // A row pointer for this lane (clamped for address safety on idle waves)
    int m = tm * 16 + lo;
    if (m >= M) m = M - 1;
    const float* Arow = A + (long)(m / rowsPerB) * aStrideB
                          + (long)(m % rowsPerB) * aStrideF;

    // Cooperative staging coordinates: thread -> (k within chunk, col in panel)
    const int skk  = tid >> 5;              // 0..7
    const int sc   = tid & 31;              // 0..31
    const int gcol = colBase + sc;
    const unsigned ldsoff = (unsigned)((skk * 32 + sc) * 4);

    v8f c0 = {}, c1 = {};

    for (int k0 = 0; k0 < K; k0 += 8) {
        __syncthreads();                    // previous chunk fully consumed
        if (gcol < N) {
            const float* gp = Bw + (long)(k0 + skk) * N + gcol;
            asm volatile("global_load_async_to_lds_b32 %0, %1, off"
                         :: "v"(ldsoff),
                            "v"((unsigned long long)(uintptr_t)gp)
                         : "memory");
        } else {
            Bs[skk][sc] = 0.f;              // zero-fill ragged columns
        }
        wait_asynccnt0();                   // drain own async loads
        __syncthreads();                    // panel visible to all waves

        if (active) {
            #pragma unroll
            for (int q = 0; q < 2; ++q) {
                const int kb = q * 4 + 2 * hi;   // reg j -> K = kb + j
                v2f a, b0;
                a.x  = Arow[k0 + kb];
                a.y  = Arow[k0 + kb + 1];
                b0.x = Bs[kb][lo];
                b0.y = Bs[kb + 1][lo];
                c0 = wmma_f32(a, b0, c0);
                if (nt1ok) {
                    v2f b1;
                    b1.x = Bs[kb][16 + lo];
                    b1.y = Bs[kb + 1][16 + lo];
                    c1 = wmma_f32(a, b1, c1);
                }
            }
        }
    }

    if (!active) return;

    // ---- epilogue: bias / relu / gate mask, batched-strided store ----
    const int col0 = colBase + lo;
    const float bv0 = bias ? bias[col0] : 0.f;
    #pragma unroll
    for (int r = 0; r < 8; ++r) {
        const int row = tm * 16 + r + 8 * hi;
        const int bb = row / rowsPerB;
        const int ff = row - bb * rowsPerB;
        float v = c0[r] + bv0;
        if (relu) v = fmaxf(v, 0.f);
        if (gates) v *= (gates[bb * NEXP + expert] > 0.f) ? 1.f : 0.f;
        C[(long)bb * cStrideB + (long)ff * cStrideF + col0] = v;
    }
    if (nt1ok) {
        const int col1 = colBase + 16 + lo;
        const float bv1 = bias ? bias[col1] : 0.f;
        #pragma unroll
        for (int r = 0; r < 8; ++r) {
            const int row = tm * 16 + r + 8 * hi;
            const int bb = row / rowsPerB;
            const int ff = row - bb * rowsPerB;
            float v = c1[r] + bv1;
            if (relu) v = fmaxf(v, 0.f);
            if (gates) v *= (gates[bb * NEXP + expert] > 0.f) ? 1.f : 0.f;
            C[(long)bb * cStrideB + (long)ff * cStrideF + col1] = v;
        }
    }
}

// ---------------------------------------------------------------------------
// Projection GEMM per expert: Z[b, t, d] = sum_s pw[s, t] * xemb[b, s, d]
//                                          + pb[t] + PE[t, d]
// ---------------------------------------------------------------------------
__global__ __launch_bounds__(256) void proj_wmma_f32(
    int sp,
    const float* __restrict__ xemb, const float* __restrict__ pw,
    const float* __restrict__ pb,  const float* __restrict__ pe,
    float* __restrict__ Z)
{
    const int lane = threadIdx.x & 31;
    const int wid  = blockIdx.x * (blockDim.x >> 5) + (threadIdx.x >> 5);
    const int tilesM = (sp + 15) >> 4;
    if (wid >= NB * tilesM) return;

    const int b  = wid / tilesM;
    const int tm = wid - b * tilesM;
    const int lo = lane & 15;
    const int hi = lane >> 4;

    const int m = tm * 16 + lo;                 // column t of pw
    const bool mok = (m < sp);
    const float* Bbase = xemb + (long)b * (SEQ * DE) + lo;

    v8f c = {};
    for (int k0 = 0; k0 < SEQ; k0 += 4) {
        const int ka = k0 + 2 * hi;
        v2f a, bv;
        a.x  = mok ? pw[(long)ka * sp + m] : 0.f;
        a.y  = mok ? pw[(long)(ka + 1) * sp + m] : 0.f;
        bv.x = Bbase[ka * DE];
        bv.y = Bbase[(ka + 1) * DE];
        c = wmma_f32(a, bv, c);
    }

    const int col = lo;
    #pragma unroll
    for (int r = 0; r < 8; ++r) {
        const int row = tm * 16 + r + 8 * hi;
        if (row < sp) {
            float v = c[r] + pb[row] + pe[row * DE + col];
            Z[(long)b * sp * DE + (long)row * DE + col] = v;
        }
    }
}

// ---------------------------------------------------------------------------
// Host-side orchestration
// ---------------------------------------------------------------------------
extern "C" void kernel_launch(void* const* d_in, const int* in_sizes, int n_in,
                              void* d_out, int out_size, void* d_ws, size_t ws_size,
                              hipStream_t stream) {
    (void)in_sizes; (void)n_in; (void)out_size; (void)ws_size;

    const float* x       = (const float*)d_in[0];
    const float* start_w = (const float*)d_in[1];
    const float* start_b = (const float*)d_in[2];
    const float* w1      = (const float*)d_in[3];
    const float* b1      = (const float*)d_in[4];
    const float* w2      = (const float*)d_in[5];
    const float* b2      = (const float*)d_in[6];
    const float* w_gate  = (const float*)d_in[7];
    const float* value_w = (const float*)d_in[8];
    const float* const* proj_ws = (const float* const*)&d_in[9];
    const float* const* proj_bs = (const float* const*)&d_in[9 + NEXP];
    const float* const* fc1_ws  = (const float* const*)&d_in[9 + 2 * NEXP];
    const float* const* fc1_bs  = (const float* const*)&d_in[9 + 3 * NEXP];
    const float* const* fc2_ws  = (const float* const*)&d_in[9 + 4 * NEXP];
    const float* const* fc2_bs  = (const float* const*)&d_in[9 + 5 * NEXP];

    static const int PS[NEXP] = {96,48,32,24,19,16,13,12,10,9,8,7,6,5,4,3,2};
    int SP[NEXP], prefix[NEXP];
    {
        int acc = 0;
        for (int i = 0; i < NEXP; ++i) {
            SP[i] = ((SEQ + PS[i] - 1) / PS[i]) * PS[i];
            prefix[i] = acc;
            acc += SP[i];
        }
    }

    float* ws    = (float*)d_ws;
    float* pe    = ws;                      // 192*16           = 3072
    float* gates = pe + 192 * DE;           // 64*17            = 1088
    float* xemb  = gates + NB * NEXP;       // 64*96*16         = 98304
    float* Z     = xemb + NB * SEQ * DE;    // <= 64*128*16     = 131072
    float* H     = Z + NB * 128 * DE;       // <= 64*128*64     = 524288
    float* out   = (float*)d_out;

    // 1) positional-encoding table
    pe_kernel<<<(192 * DE + 255) / 256, 256, 0, stream>>>(pe);

    // 2) gates (also written to output tail)
    gating_kernel<<<NB, 96, 0, stream>>>(x, start_w, start_b, w1, b1, w2, b2,
                                         w_gate, gates, out + OUT_MAIN);

    // 3) embedding: xemb[B*S, 16] = x[B*S, 32] @ value_w[32, 16]
    {
        const int M = NB * SEQ;             // 6144
        const int tiles = (M / 16) * (DE / 16);
        gemm_wmma_nt1<<<(tiles + 7) / 8, 256, 0, stream>>>(
            M, DE, 32,
            x, (long)0, (long)32, M,
            value_w, nullptr,
            xemb, (long)0, (long)DE,
            nullptr, 0, 0);
    }

    // 4) per-expert: proj(+PE) -> fc1(relu) -> fc2(mask) -> d_out slice
    for (int i = 0; i < NEXP; ++i) {
        const int p   = PS[i];
        const int sp  = SP[i];
        const int F   = sp / p;
        const int din = p * DE;
        const int dh  = din * 4;
        const int Mi  = NB * F;
        const int tilesM = Mi / 16;
        const int mPanels = (tilesM + 7) / 8;

        // proj: Z[b, sp, 16]
        {
            const int tM = (sp + 15) / 16;
            const int waves = NB * tM;
            proj_wmma_f32<<<(waves + 7) / 8, 256, 0, stream>>>(
                sp, xemb, proj_ws[i], proj_bs[i], pe, Z);
        }
        // fc1: H[Mi, dh] = relu(Z2[Mi, din] @ W1 + b1)
        {
            const int panelsN = (dh / 16 + 1) / 2;
            gemm_wmma_nt2<<<panelsN * mPanels, 256, 0, stream>>>(
                Mi, dh, din,
                Z, (long)sp * DE, (long)din, F,
                fc1_ws[i], fc1_bs[i],
                H, (long)F * dh, (long)dh,
                nullptr, 0, 1, panelsN);
        }
        // fc2: out[b, prefix*16 + f*din + n] = (H @ W2 + b2) * mask(b, i)
        {
            const int panelsN = (din / 16 + 1) / 2;
            gemm_wmma_nt2<<<panelsN * mPanels, 256, 0, stream>>>(
                Mi, din, dh,
                H, (long)F * dh, (long)dh, F,
                fc2_ws[i], fc2_bs[i],
                out + (long)DE * prefix[i], (long)OUT_STRIDE, (long)din,
                gates, i, 0, panelsN);
        }
    }
}